// MultiEnvLight_2534030704856
// MI455X (gfx1250) — compile-verified
//
#include <hip/hip_runtime.h>
#include <math.h>

// MultiEnvLight "pure_env" forward for MI455X (gfx1250).
// Memory-gather-bound: 32 random RGB texel reads per query from a 301MB table
// (~1-2 GB of line traffic total -> ~45-90us floor at 23.3 TB/s; FLOPs are
// negligible). Strategy: 1 thread = 1 query, phase-ordered so all 16
// global_prefetch_b8 issue before the floating-point weight math (real overlap
// work), then a fully unrolled gather so the wave keeps 32 independent
// global_load_b96 in flight behind the gfx1250 split load counters.

#define RES 128
#define FACE_TEXELS (RES * RES)          // 16384
#define PROBE_TEXELS (6 * FACE_TEXELS)   // 98304

__global__ __launch_bounds__(256) void envlight_kernel(
    const float* __restrict__ xyz,
    const float* __restrict__ l,
    const float* __restrict__ base,
    float* __restrict__ out,
    int n)
{
    int i = blockIdx.x * blockDim.x + threadIdx.x;
    if (i >= n) return;

    // ================= Phase 1: integer address generation ==================
    // direction -> cube face/uv (nvdiffrast convention)
    float lx = l[3 * i + 0], ly = l[3 * i + 1], lz = l[3 * i + 2];
    float nrm = sqrtf(lx * lx + ly * ly + lz * lz);
    float inv = 1.0f / fmaxf(nrm, 1e-9f);
    float dx = lx * inv, dy = ly * inv, dz = lz * inv;

    float ax = fabsf(dx), ay = fabsf(dy), az = fabsf(dz);
    bool is_x = (ax >= ay) && (ax >= az);
    bool is_y = (!is_x) && (ay >= az);

    int face;
    float ma, sc, tc;
    if (is_x) {
        face = (dx >= 0.0f) ? 0 : 1;
        ma = ax;
        sc = (dx >= 0.0f) ? -dz : dz;
        tc = -dy;
    } else if (is_y) {
        face = (dy >= 0.0f) ? 2 : 3;
        ma = ay;
        sc = dx;
        tc = (dy >= 0.0f) ? dz : -dz;
    } else {
        face = (dz >= 0.0f) ? 4 : 5;
        ma = az;
        sc = (dz >= 0.0f) ? dx : -dx;
        tc = -dy;
    }
    ma = fmaxf(ma, 1e-9f);
    float u = 0.5f * (sc / ma + 1.0f);
    float v = 0.5f * (tc / ma + 1.0f);

    float fx = u * (float)RES - 0.5f;
    float fy = v * (float)RES - 0.5f;
    float x0f = floorf(fx), y0f = floorf(fy);
    float tx = fx - x0f, ty = fy - y0f;
    int x0 = min(max((int)x0f, 0), RES - 1);
    int x1 = min(max((int)x0f + 1, 0), RES - 1);
    int y0 = min(max((int)y0f, 0), RES - 1);
    int y1 = min(max((int)y0f + 1, 0), RES - 1);

    int tap[4];
    tap[0] = y0 * RES + x0;
    tap[1] = y0 * RES + x1;
    tap[2] = y1 * RES + x0;
    tap[3] = y1 * RES + x1;

    // trilinear probe corner indices (grid 8x8x4)
    float c0 = fminf(fmaxf(xyz[3 * i + 0] * 7.0f, 0.0f), 7.0f);
    float c1 = fminf(fmaxf(xyz[3 * i + 1] * 7.0f, 0.0f), 7.0f);
    float c2 = fminf(fmaxf(xyz[3 * i + 2] * 3.0f, 0.0f), 3.0f);
    int i0x = (int)floorf(c0), i0y = (int)floorf(c1), i0z = (int)floorf(c2);
    int i1x = min(i0x + 1, 7), i1y = min(i0y + 1, 7), i1z = min(i0z + 1, 3);
    float t0 = c0 - (float)i0x, t1 = c1 - (float)i0y, t2 = c2 - (float)i0z;

    int poff[8];  // texel offset of each probe-face
#pragma unroll
    for (int p = 0; p < 8; ++p) {
        int ix = ((p >> 2) & 1) ? i1x : i0x;
        int iy = ((p >> 1) & 1) ? i1y : i0y;
        int iz = (p & 1) ? i1z : i0z;
        poff[p] = (ix * 32 + iy * 4 + iz) * PROBE_TEXELS + face * FACE_TEXELS;
    }

    // ================= Phase 2: prefetch every bilinear footprint ==========
    // gfx1250 global_prefetch_b8: warm both texel rows of all 8 probe-faces
    // before any dependent use; phase 3's VALU work overlaps the fills.
#pragma unroll
    for (int p = 0; p < 8; ++p) {
        const float* pb = base + (size_t)poff[p] * 3;
        __builtin_prefetch(pb + (size_t)tap[0] * 3, 0, 3);
        __builtin_prefetch(pb + (size_t)tap[2] * 3, 0, 3);
    }

    // ================= Phase 3: floating-point weight math =================
    float tapw[4];
    tapw[0] = (1.0f - ty) * (1.0f - tx);
    tapw[1] = (1.0f - ty) * tx;
    tapw[2] = ty * (1.0f - tx);
    tapw[3] = ty * tx;

    float pw[8];
    float wsum = 0.0f;
#pragma unroll
    for (int p = 0; p < 8; ++p) {
        float w = (((p >> 2) & 1) ? t0 : 1.0f - t0) *
                  (((p >> 1) & 1) ? t1 : 1.0f - t1) *
                  ((p & 1) ? t2 : 1.0f - t2);
        pw[p] = w;
        wsum += w;
    }
    float winv = 1.0f / fmaxf(wsum, 1e-8f);

    // ================= Phase 4: gather + two-level weighted blend ==========
    float r = 0.0f, g = 0.0f, b = 0.0f;
#pragma unroll
    for (int p = 0; p < 8; ++p) {
        const float* pb = base + (size_t)poff[p] * 3;
        float br = 0.0f, bg = 0.0f, bb = 0.0f;
#pragma unroll
        for (int t = 0; t < 4; ++t) {
            const float* tp = pb + (size_t)tap[t] * 3;  // one b96 per texel
            float w = tapw[t];
            br = fmaf(w, tp[0], br);
            bg = fmaf(w, tp[1], bg);
            bb = fmaf(w, tp[2], bb);
        }
        float w = pw[p] * winv;
        r = fmaf(w, br, r);
        g = fmaf(w, bg, g);
        b = fmaf(w, bb, b);
    }

    // ================= sigmoid * 10 ========================================
    out[3 * i + 0] = 10.0f / (1.0f + expf(-r));
    out[3 * i + 1] = 10.0f / (1.0f + expf(-g));
    out[3 * i + 2] = 10.0f / (1.0f + expf(-b));
}

extern "C" void kernel_launch(void* const* d_in, const int* in_sizes, int n_in,
                              void* d_out, int out_size, void* d_ws, size_t ws_size,
                              hipStream_t stream) {
    const float* xyz  = (const float*)d_in[0];
    const float* l    = (const float*)d_in[1];
    const float* base = (const float*)d_in[2];
    float* out = (float*)d_out;
    int n = in_sizes[0] / 3;  // 524288 queries
    dim3 block(256);          // 8 wave32s per workgroup
    dim3 grid((n + 255) / 256);
    envlight_kernel<<<grid, block, 0, stream>>>(xyz, l, base, out, n);
}